// GCNEncoder_22273700397754
// MI455X (gfx1250) — compile-verified
//
#include <hip/hip_runtime.h>

#define D_MODEL 128

typedef __attribute__((ext_vector_type(2))) float v2f;
typedef __attribute__((ext_vector_type(8))) float v8f;

// ---------------------------------------------------------------------------
// Degree accumulation: deg[dst] += 1 per edge (self-loops handled in rsqrt).
// ---------------------------------------------------------------------------
__global__ void degree_kernel(const long long* __restrict__ dst,
                              float* __restrict__ deg, int E) {
    int e = blockIdx.x * blockDim.x + threadIdx.x;
    if (e < E) atomicAdd(&deg[(int)dst[e]], 1.0f);
}

// dis[i] = rsqrt(deg[i] + 1)   (+1 = self loop; always > 0)
__global__ void rsqrt_kernel(float* __restrict__ d, int N) {
    int i = blockIdx.x * blockDim.x + threadIdx.x;
    if (i < N) d[i] = rsqrtf(d[i] + 1.0f);
}

// ---------------------------------------------------------------------------
// out[N x 128] = A[N x K] @ W[K x 128] with V_WMMA_F32_16X16X4_F32.
// Block = 256 threads = 8 wave32; block covers 16 rows, wave w covers cols
// [16w, 16w+16). A stripe staged in LDS (stride 132 -> conflict-free reads).
// ---------------------------------------------------------------------------
__global__ __launch_bounds__(256) void gemm_wmma_f32(
    const float* __restrict__ A, const float* __restrict__ W,
    float* __restrict__ out, int nrows, int K)
{
    constexpr int LDA = 132;                 // pad: lane m hits banks 4m+kk
    __shared__ float ldsA[16 * LDA];
    const int tid  = threadIdx.x;
    const int row0 = blockIdx.x * 16;

    // Cooperative, coalesced load of the 16 x K A-stripe into LDS.
    for (int idx = tid; idx < 16 * K; idx += 256) {
        int r = idx / K, c = idx - r * K;
        int gr = row0 + r; if (gr >= nrows) gr = nrows - 1;   // clamp (no EXEC divergence later)
        ldsA[r * LDA + c] = A[(size_t)gr * K + c];
    }
    __syncthreads();

    const int wave  = tid >> 5;              // 0..7  -> column tile
    const int lane  = tid & 31;
    const int col0  = wave * 16;
    const int m     = lane & 15;             // M (for A) / N (for B) index
    const int khalf = (lane >> 4) << 1;      // lanes 16..31 hold K = 2,3

    v8f c = {};
    for (int k0 = 0; k0 < K; k0 += 4) {
        const int kk = k0 + khalf;
        v2f a, b;
        // A 16x4 fragment: lane m holds A[m][kk], A[m][kk+1]
        a.x = ldsA[m * LDA + kk];
        a.y = ldsA[m * LDA + kk + 1];
        // B 4x16 fragment: lane holds W[kk][col0+m], W[kk+1][col0+m]
        b.x = W[(size_t)kk       * D_MODEL + col0 + m];
        b.y = W[(size_t)(kk + 1) * D_MODEL + col0 + m];
        c = __builtin_amdgcn_wmma_f32_16x16x4_f32(
                /*neg_a=*/false, a, /*neg_b=*/false, b,
                /*c_mod=*/(short)0, c, /*reuse_a=*/false, /*reuse_b=*/false);
    }

    // C/D layout: VGPR r, lanes 0-15 -> M=r, lanes 16-31 -> M=r+8; N = lane&15
    const int nidx  = col0 + m;
    const int rbase = row0 + ((lane >> 4) << 3);
#pragma unroll
    for (int r = 0; r < 8; ++r) {
        int gr = rbase + r;
        if (gr < nrows) out[(size_t)gr * D_MODEL + nidx] = c[r];
    }
}

// ---------------------------------------------------------------------------
// Edge scatter: one wave per edge; lane handles 4 of the 128 features.
// acc[dst] += h[src] * dis[src] * dis[dst]
// ---------------------------------------------------------------------------
__global__ __launch_bounds__(256) void scatter_kernel(
    const float* __restrict__ h, const long long* __restrict__ src,
    const long long* __restrict__ dst, const float* __restrict__ dis,
    float* __restrict__ acc, int E)
{
    long long t = (long long)blockIdx.x * 256 + threadIdx.x;
    int e    = (int)(t >> 5);
    int lane = (int)(t & 31);
    if (e >= E) return;
    int s = (int)src[e];
    int d = (int)dst[e];
    float norm = dis[s] * dis[d];
    const float4 v = *(const float4*)(h + (size_t)s * D_MODEL + lane * 4);
    float* o = acc + (size_t)d * D_MODEL + lane * 4;
    atomicAdd(o + 0, v.x * norm);
    atomicAdd(o + 1, v.y * norm);
    atomicAdd(o + 2, v.z * norm);
    atomicAdd(o + 3, v.w * norm);
}

// ---------------------------------------------------------------------------
// out = [relu](acc + hw * dis^2 + bias)   (self-loop folded in, no atomics)
// One thread per 4 features.
// ---------------------------------------------------------------------------
__global__ __launch_bounds__(256) void finalize_kernel(
    const float* __restrict__ acc, const float* __restrict__ hw,
    const float* __restrict__ dis, const float* __restrict__ bias,
    float* __restrict__ out, int N, int do_relu)
{
    long long t = (long long)blockIdx.x * 256 + threadIdx.x;
    int i    = (int)(t >> 5);
    int lane = (int)(t & 31);
    if (i >= N) return;
    float d2 = dis[i]; d2 *= d2;
    size_t off = (size_t)i * D_MODEL + lane * 4;
    float4 a = *(const float4*)(acc + off);
    float4 h = *(const float4*)(hw + off);
    float4 b = *(const float4*)(bias + lane * 4);
    float4 r;
    r.x = a.x + h.x * d2 + b.x;
    r.y = a.y + h.y * d2 + b.y;
    r.z = a.z + h.z * d2 + b.z;
    r.w = a.w + h.w * d2 + b.w;
    if (do_relu) {
        r.x = fmaxf(r.x, 0.0f); r.y = fmaxf(r.y, 0.0f);
        r.z = fmaxf(r.z, 0.0f); r.w = fmaxf(r.w, 0.0f);
    }
    *(float4*)(out + off) = r;
}

// ---------------------------------------------------------------------------
extern "C" void kernel_launch(void* const* d_in, const int* in_sizes, int n_in,
                              void* d_out, int out_size, void* d_ws, size_t ws_size,
                              hipStream_t stream)
{
    const float*     x  = (const float*)d_in[0];
    const long long* ei = (const long long*)d_in[1];   // int64 (2, E)
    const float*     W1 = (const float*)d_in[2];
    const float*     b1 = (const float*)d_in[3];
    const float*     W2 = (const float*)d_in[4];
    const float*     b2 = (const float*)d_in[5];
    const float*     W3 = (const float*)d_in[6];
    const float*     b3 = (const float*)d_in[7];

    const int K1 = in_sizes[2] / D_MODEL;     // 4
    const int N  = in_sizes[0] / K1;          // 100000
    const int E  = in_sizes[1] / 2;           // 1600000
    const long long* srcp = ei;
    const long long* dstp = ei + E;

    // workspace partition (256B-aligned)
    char* ws = (char*)d_ws;
    size_t featBytes = ((size_t)N * D_MODEL * sizeof(float) + 255) & ~(size_t)255;
    size_t disBytes  = (((size_t)N * sizeof(float)) + 255) & ~(size_t)255;
    float* dis = (float*)ws;
    float* hw  = (float*)(ws + disBytes);
    float* acc = (float*)(ws + disBytes + featBytes);
    float* h1  = (float*)(ws + disBytes + 2 * featBytes);

    const int nodeBlks  = (N + 255) / 256;
    const int edgeBlks  = (E + 255) / 256;
    const int gemmBlks  = (N + 15) / 16;
    const long long edgeWaveBlks = ((long long)E * 32 + 255) / 256;
    const long long nodeWaveBlks = ((long long)N * 32 + 255) / 256;

    // normalization: deg -> dis = rsqrt(deg+1)
    hipMemsetAsync(dis, 0, (size_t)N * sizeof(float), stream);
    degree_kernel<<<edgeBlks, 256, 0, stream>>>(dstp, dis, E);
    rsqrt_kernel<<<nodeBlks, 256, 0, stream>>>(dis, N);

    // ---- layer 1 ----
    gemm_wmma_f32<<<gemmBlks, 256, 0, stream>>>(x, W1, hw, N, K1);
    hipMemsetAsync(acc, 0, (size_t)N * D_MODEL * sizeof(float), stream);
    scatter_kernel<<<(unsigned)edgeWaveBlks, 256, 0, stream>>>(hw, srcp, dstp, dis, acc, E);
    finalize_kernel<<<(unsigned)nodeWaveBlks, 256, 0, stream>>>(acc, hw, dis, b1, h1, N, 1);

    // ---- layer 2 ----
    gemm_wmma_f32<<<gemmBlks, 256, 0, stream>>>(h1, W2, hw, N, D_MODEL);
    hipMemsetAsync(acc, 0, (size_t)N * D_MODEL * sizeof(float), stream);
    scatter_kernel<<<(unsigned)edgeWaveBlks, 256, 0, stream>>>(hw, srcp, dstp, dis, acc, E);
    finalize_kernel<<<(unsigned)nodeWaveBlks, 256, 0, stream>>>(acc, hw, dis, b2, h1, N, 1);

    // ---- layer 3 ----
    gemm_wmma_f32<<<gemmBlks, 256, 0, stream>>>(h1, W3, hw, N, D_MODEL);
    hipMemsetAsync(acc, 0, (size_t)N * D_MODEL * sizeof(float), stream);
    scatter_kernel<<<(unsigned)edgeWaveBlks, 256, 0, stream>>>(hw, srcp, dstp, dis, acc, E);
    finalize_kernel<<<(unsigned)nodeWaveBlks, 256, 0, stream>>>(acc, hw, dis, b3, (float*)d_out, N, 0);
}